// diff_voxelize_6253472383906
// MI455X (gfx1250) — compile-verified
//
#include <hip/hip_runtime.h>
#include <hip/hip_bf16.h>

// MI455X (gfx1250) fused differentiable voxelization.
//  K0: zero voxel workspace (134 MB, grid-stride float4)
//  K1: trilinear splat, 8 relaxed agent-scope float atomics per point
//  K2: fused 3x3x3 separable Gaussian:
//      - stage 64 halo z-lines/WG into LDS (GLOBAL_LOAD_ASYNC_TO_LDS_B128,
//        512B coalesced bursts per wave, ASYNCcnt-tracked)
//      - z-axis as block-tridiagonal matmul on V_WMMA_F32_16X16X4_F32,
//        all A-chunks preloaded to registers -> 24 back-to-back WMMAs
//      - zconv written in-place over the raw LDS tile, y/x via 9-tap VALU
//      One volume read + one write => ~268 MB HBM ~= 11.5 us at 23.3 TB/s.

typedef __attribute__((ext_vector_type(2))) float v2f;
typedef __attribute__((ext_vector_type(8))) float v8f;

#define DIM 128
#define TS  132   // padded LDS z-stride -> conflict-free for all access phases

#if __has_builtin(__builtin_amdgcn_global_load_async_to_lds_b128)
#define HAVE_ASYNC_LDS 1
// Param types per hipcc diagnostic: global side is "v4i __device__ *"
// (address_space(1) pointer to int32x4); LDS side mirrored in AS(3).
typedef __attribute__((__vector_size__(4 * sizeof(int)))) int v4i;
typedef __attribute__((address_space(1))) v4i GA4;   // global  (__device__)
typedef __attribute__((address_space(3))) v4i LA4;   // LDS     (__shared__)
#endif

static __device__ __forceinline__ void wait_async_lds() {
#if defined(HAVE_ASYNC_LDS)
#if __has_builtin(__builtin_amdgcn_s_wait_asynccnt)
    __builtin_amdgcn_s_wait_asynccnt(0);
#else
    asm volatile("s_wait_asynccnt 0x0" ::: "memory");
#endif
#endif
}

// ---------------------------------------------------------------- zero ----
__global__ void vox_zero_kernel(float4* __restrict__ p, int n4) {
    int i = blockIdx.x * blockDim.x + threadIdx.x;
    int stride = gridDim.x * blockDim.x;
    float4 z = make_float4(0.f, 0.f, 0.f, 0.f);
    for (; i < n4; i += stride) p[i] = z;
}

// --------------------------------------------------------------- splat ----
__global__ void splat_kernel(const float* __restrict__ pc,
                             float* __restrict__ vox,
                             int npts, long long total) {
    long long p = (long long)blockIdx.x * blockDim.x + threadIdx.x;
    if (p >= total) return;
    int b = (int)(p / npts);
    int i = (int)(p % npts);
    const float* q = pc + ((long long)b * npts + i) * 3;
    const float EPS = 1e-6f;
    float nx = q[0] * (1.f / DIM) - 0.5f;
    float ny = q[1] * (1.f / DIM) - 0.5f;
    float nz = q[2] * (1.f / DIM) - 0.5f;
    bool valid = (nx < 0.5f - EPS) && (nx > -0.5f + EPS) &&
                 (ny < 0.5f - EPS) && (ny > -0.5f + EPS) &&
                 (nz < 0.5f - EPS) && (nz > -0.5f + EPS);
    if (!valid) return;
    float gx = (nx + 0.5f) * (DIM - 1.f);
    float gy = (ny + 0.5f) * (DIM - 1.f);
    float gz = (nz + 0.5f) * (DIM - 1.f);
    float fx = floorf(gx), fy = floorf(gy), fz = floorf(gz);
    int ix = (int)fx, iy = (int)fy, iz = (int)fz;
    float rx = gx - fx, ry = gy - fy, rz = gz - fz;
    float wx[2] = {1.f - rx, rx};
    float wy[2] = {1.f - ry, ry};
    float wz[2] = {1.f - rz, rz};
#pragma unroll
    for (int dx = 0; dx < 2; ++dx)
#pragma unroll
        for (int dy = 0; dy < 2; ++dy)
#pragma unroll
            for (int dz = 0; dz < 2; ++dz) {
                float w = wx[dx] * wy[dy] * wz[dz];
                int idx = (((b * DIM + ix + dx) * DIM) + iy + dy) * DIM + iz + dz;
                (void)__hip_atomic_fetch_add(&vox[idx], w, __ATOMIC_RELAXED,
                                             __HIP_MEMORY_SCOPE_AGENT);
            }
}

// -------------------------------------------------------------- smooth ----
// Tile: output (2 x 14 x 128); input lines (lx=4) x (ly=16), full z + halo.
// Wave w (0..7): line-group g = w>>1 (lx), z-half jb = (w&1)*4.
__global__ __launch_bounds__(256) void smooth_kernel(const float* __restrict__ vox,
                                                     const float* __restrict__ sigp,
                                                     float* __restrict__ out) {
    __shared__ __align__(16) float zc[64 * TS];   // raw lines, then zconv in place

    const int tid  = threadIdx.x;
    const int lane = tid & 31;
    const int wv   = tid >> 5;
    const int g    = wv >> 1;        // input x-line group 0..3
    const int jb   = (wv & 1) * 4;   // z-tile half

    const int x0 = blockIdx.x * 2;   // 64 x-tiles (exact)
    const int y0 = blockIdx.y * 14;  // 10 y-tiles (last partial)
    const int b  = blockIdx.z;

    // Gaussian taps: kern = [e,1,e]/(2e+1), e = exp(-1/(2 sigma^2))
    float sigma = sigp[0];
    float e  = __expf(-1.f / (2.f * sigma * sigma));
    float nf = 1.f / (2.f * e + 1.f);
    float kw = e * nf, kc = nf;

    // --- stage 64 raw lines into LDS: one line per wave per iter,
    //     lane k loads float4 at z=4k -> 512B contiguous per wave ---------
    for (int k = 0; k < 8; ++k) {
        int line = k * 8 + wv;
        int lx = line >> 4, ly = line & 15;
        int gxx = x0 - 1 + lx, gyy = y0 - 1 + ly;
        bool v = (gxx >= 0) & (gxx < DIM) & (gyy >= 0) & (gyy < DIM);
        float* dst = &zc[line * TS + 4 * lane];
        if (v) {
            const float* src = vox + (((long long)b * DIM + gxx) * DIM + gyy) * DIM
                               + 4 * lane;
#if defined(HAVE_ASYNC_LDS)
            __builtin_amdgcn_global_load_async_to_lds_b128((GA4*)src, (LA4*)dst, 0, 0);
#else
            *(float4*)dst = *(const float4*)src;
#endif
        } else {
            *(float4*)dst = make_float4(0.f, 0.f, 0.f, 0.f);
        }
    }
    wait_async_lds();
    __syncthreads();

    // --- preload this lane's 18 A-chunks (branchless, clipped) -----------
    const int N    = lane & 15;               // matrix row M / column N
    const int zoff = (lane < 16) ? 0 : 2;     // A layout K split
    const int myln = g * 16 + N;
    v2f A[18];                                // m=0..17 <-> z = 16*jb + 4*(m-1) + zoff
#pragma unroll
    for (int m = 0; m < 18; ++m) {
        int z = 16 * jb + 4 * (m - 1) + zoff;
        bool inr = (unsigned)z < (unsigned)DIM;
        int zs = inr ? z : 0;
        float a0 = zc[myln * TS + zs];
        float a1 = zc[myln * TS + zs + 1];
        A[m].x = inr ? fminf(fmaxf(a0, 0.f), 1.f) : 0.f;   // clip(vox,0,1)
        A[m].y = inr ? fminf(fmaxf(a1, 0.f), 1.f) : 0.f;
    }
    __syncthreads();   // raw tile fully consumed -> zc reusable for zconv

    // --- B operands (lane-resident), ISA B 4x16 layout mirrored from A ---
    const int Kb = (lane < 16) ? 0 : 2;
    v2f bT[4];
#pragma unroll
    for (int q = 0; q < 4; ++q) {
        int d0 = 4 * q + Kb - N;
        int d1 = d0 + 1;
        bT[q].x = (d0 == 0) ? kc : ((d0 == 1 || d0 == -1) ? kw : 0.f);
        bT[q].y = (d1 == 0) ? kc : ((d1 == 1 || d1 == -1) ? kw : 0.f);
    }
    v2f bEp; bEp.x = 0.f;                      bEp.y = (lane == 16) ? kw : 0.f; // (K=15,N=0)
    v2f bEn; bEn.x = (lane == 15) ? kw : 0.f;  bEn.y = 0.f;                     // (K=0,N=15)

    // --- z pass: out_tile_j = A_j*T + A_{j-1}*E- + A_{j+1}*E+ -------------
    const int mhi = (lane < 16) ? 0 : 8;
#pragma unroll
    for (int jj = 0; jj < 4; ++jj) {
        v8f c = {0.f, 0.f, 0.f, 0.f, 0.f, 0.f, 0.f, 0.f};
#pragma unroll
        for (int q = 0; q < 4; ++q)
            c = __builtin_amdgcn_wmma_f32_16x16x4_f32(false, A[1 + 4 * jj + q],
                                                      false, bT[q],
                                                      (short)0, c, false, false);
        c = __builtin_amdgcn_wmma_f32_16x16x4_f32(false, A[4 * jj],     false, bEp,
                                                  (short)0, c, false, false);
        c = __builtin_amdgcn_wmma_f32_16x16x4_f32(false, A[4 * jj + 5], false, bEn,
                                                  (short)0, c, false, false);

        // scatter C: vgpr p holds M = p (+8 for upper half-wave), N = z-in-tile
        const int zz = 16 * (jb + jj) + N;
#pragma unroll
        for (int p = 0; p < 8; ++p)
            zc[(g * 16 + p + mhi) * TS + zz] = c[p];
    }
    __syncthreads();

    // --- y/x pass: 9-tap from LDS, coalesced clipped stores ---------------
    const int z = tid & 127;
    const int h = tid >> 7;
    float wgt[3] = {kw, kc, kw};
#pragma unroll
    for (int ox = 0; ox < 2; ++ox) {
        const int xg = x0 + ox;
        for (int t = 0; t < 7; ++t) {
            const int oy = h * 7 + t;
            const int yg = y0 + oy;
            if (yg >= DIM) continue;
            float acc = 0.f;
#pragma unroll
            for (int dx = 0; dx < 3; ++dx)
#pragma unroll
                for (int dy = 0; dy < 3; ++dy)
                    acc += wgt[dx] * wgt[dy] * zc[((ox + dx) * 16 + (oy + dy)) * TS + z];
            acc = fminf(fmaxf(acc, 0.f), 1.f);
            out[(((long long)b * DIM + xg) * DIM + yg) * DIM + z] = acc;
        }
    }
}

// -------------------------------------------------------------- launch ----
extern "C" void kernel_launch(void* const* d_in, const int* in_sizes, int n_in,
                              void* d_out, int out_size, void* d_ws, size_t ws_size,
                              hipStream_t stream) {
    const float* pc  = (const float*)d_in[0];
    const float* sig = (const float*)d_in[1];
    float* out = (float*)d_out;
    float* vox = (float*)d_ws;

    const int vol  = DIM * DIM * DIM;            // 2,097,152
    const int bs   = out_size / vol;             // 16
    const int npts = in_sizes[0] / (bs * 3);     // 100,000

    const int n4 = (bs * vol) / 4;
    vox_zero_kernel<<<2048, 256, 0, stream>>>((float4*)vox, n4);

    const long long total = (long long)bs * npts;
    const int blocks = (int)((total + 255) / 256);
    splat_kernel<<<blocks, 256, 0, stream>>>(pc, vox, npts, total);

    smooth_kernel<<<dim3(DIM / 2, (DIM + 13) / 14, bs), 256, 0, stream>>>(vox, sig, out);
}